// GlobalContext_33131377721805
// MI455X (gfx1250) — compile-verified
//
#include <hip/hip_runtime.h>
#include <hip/hip_bf16.h>

typedef __attribute__((ext_vector_type(2))) float v2f;
typedef __attribute__((ext_vector_type(8))) float v8f;

#define NROW 1024
#define NCH  256

// ---------------------------------------------------------------------------
// Kernel 1: s[i] = dot(x[i,:], w)  via V_WMMA_F32_16X16X4_F32 (fp32 in/out).
// One wave (32 threads) per 16-row tile. B has w[k] replicated across all 16
// columns, so D[m][n] = s[tile*16+m] for every n. A and B are packed with the
// SAME (lane-half, vgpr)->K convention (doc: VGPR0 holds K=0 / K=2 for lane
// halves, VGPR1 holds K=1 / K=3), which makes the dot product invariant to
// the exact hardware K permutation.
// ---------------------------------------------------------------------------
__global__ void gc_s_kernel(const float* __restrict__ x,
                            const float* __restrict__ w,
                            float* __restrict__ s) {
    const int lane  = threadIdx.x;          // 0..31, full wave32
    const int tile  = blockIdx.x;           // 64 tiles of 16 rows
    const int m     = lane & 15;
    const int row   = tile * 16 + m;
    const int khalf = (lane >> 4) * 2;      // 0 for lanes 0-15, 2 for 16-31

    const float* xr = x + row * NCH;

    v8f acc = {};
    for (int k0 = 0; k0 < NCH; k0 += 4) {
        v2f a, bm;
        a.x  = xr[k0 + khalf];
        a.y  = xr[k0 + khalf + 1];
        bm.x = w[k0 + khalf];               // replicated over all 16 B columns
        bm.y = w[k0 + khalf + 1];
        acc = __builtin_amdgcn_wmma_f32_16x16x4_f32(
            /*neg_a=*/false, a, /*neg_b=*/false, bm,
            /*c_mod=*/(short)0, acc, /*reuse_a=*/false, /*reuse_b=*/false);
    }

    // D layout (16x16 f32): VGPR r holds M=r (lanes 0-15) and M=r+8 (lanes
    // 16-31); value is constant across N, so lane 0 / lane 16 carry all rows.
    if (lane == 0) {
#pragma unroll
        for (int r = 0; r < 8; ++r) s[tile * 16 + r] = acc[r];
    }
    if (lane == 16) {
#pragma unroll
        for (int r = 0; r < 8; ++r) s[tile * 16 + 8 + r] = acc[r];
    }
}

// ---------------------------------------------------------------------------
// Kernel 2: T[c] = sum_i x[i,c];  U[c] = sum_i x[i,c]*s[i];  S = sum_i s[i].
// One block of 256 threads; thread c walks column c (coalesced across c).
// ---------------------------------------------------------------------------
__global__ void gc_reduce_kernel(const float* __restrict__ x,
                                 const float* __restrict__ s,
                                 float* __restrict__ T,
                                 float* __restrict__ U,
                                 float* __restrict__ Sout) {
    const int c = threadIdx.x;              // 0..255
    float t = 0.0f, u = 0.0f;
    for (int i = 0; i < NROW; ++i) {
        const float xv = x[i * NCH + c];    // consecutive c -> consecutive addr
        const float sv = s[i];              // uniform broadcast
        t += xv;
        u += xv * sv;
    }
    T[c] = t;
    U[c] = u;

    // S = sum of all 1024 s values: 4 per thread, then LDS tree reduce.
    __shared__ float red[NCH];
    red[c] = s[c] + s[c + 256] + s[c + 512] + s[c + 768];
    __syncthreads();
    for (int off = 128; off > 0; off >>= 1) {
        if (c < off) red[c] += red[c + off];
        __syncthreads();
    }
    if (c == 0) Sout[0] = red[0];
}

// ---------------------------------------------------------------------------
// Kernel 3: out[i,c] = x[i,c]*(N*s_i + S + N*b) + (s_i + b)*T[c] + U[c]
// ---------------------------------------------------------------------------
__global__ void gc_finalize_kernel(const float* __restrict__ x,
                                   const float* __restrict__ s,
                                   const float* __restrict__ T,
                                   const float* __restrict__ U,
                                   const float* __restrict__ Sp,
                                   const float* __restrict__ b,
                                   float* __restrict__ out) {
    const int i = blockIdx.x;               // row
    const int c = threadIdx.x;              // channel
    const float si    = s[i];
    const float S     = Sp[0];
    const float bb    = b[0];
    const float alpha = (float)NROW * si + S + (float)NROW * bb;
    const float beta  = si + bb;
    const int idx = i * NCH + c;
    out[idx] = x[idx] * alpha + beta * T[c] + U[c];
}

extern "C" void kernel_launch(void* const* d_in, const int* in_sizes, int n_in,
                              void* d_out, int out_size, void* d_ws, size_t ws_size,
                              hipStream_t stream) {
    const float* x = (const float*)d_in[0];   // [1024, 256]
    const float* w = (const float*)d_in[1];   // [256]
    const float* b = (const float*)d_in[2];   // [1]
    float* out = (float*)d_out;               // [1024, 256]

    float* ws = (float*)d_ws;
    float* s_buf = ws;                        // 1024 floats
    float* T_buf = ws + 1024;                 // 256 floats
    float* U_buf = ws + 1024 + 256;           // 256 floats
    float* S_buf = ws + 1024 + 512;           // 1 float

    gc_s_kernel<<<NROW / 16, 32, 0, stream>>>(x, w, s_buf);
    gc_reduce_kernel<<<1, NCH, 0, stream>>>(x, s_buf, T_buf, U_buf, S_buf);
    gc_finalize_kernel<<<NROW, NCH, 0, stream>>>(x, s_buf, T_buf, U_buf,
                                                 S_buf, b, out);
}